// ImageSimilarityLoss_75582834475353
// MI455X (gfx1250) — compile-verified
//
#include <hip/hip_runtime.h>
#include <hip/hip_bf16.h>

typedef float v2f __attribute__((ext_vector_type(2)));
typedef float v8f __attribute__((ext_vector_type(8)));

#define IMG_W 512
#define IMG_H 512
#define N_BLOCKS 6144          // 49152 waves / 8 waves per block
#define SSIM_C1 1.0e-4f
#define SSIM_C2 9.0e-4f
#define N_ELEMS 12582912.0     // 16*3*512*512

static __device__ __forceinline__ v8f wmma_f32(v2f a, v2f b, v8f c) {
  // D(16x16,f32) = A(16x4,f32) * B(4x16,f32) + C
  return __builtin_amdgcn_wmma_f32_16x16x4_f32(false, a, false, b, (short)0, c,
                                               false, false);
}

// lane <- lane^16 half-swap, pure VALU (v_permlanex16_b32 with identity
// lane-select) instead of ds_bpermute: no DScnt waits, co-executes with WMMA.
static __device__ __forceinline__ float xswap(float v) {
  int i = __builtin_amdgcn_permlanex16(__float_as_int(v), __float_as_int(v),
                                       0x76543210, 0x76543210, false, false);
  return __int_as_float(i);
}

// Normalized 1D gaussian tap w[d], d in [0,10], else 0.  sigma=1.5.
static __device__ __forceinline__ float gband(int d, float invS) {
  if (d < 0 || d > 10) return 0.0f;
  float t = (float)(d - 5);
  return __expf(-t * t * (1.0f / 4.5f)) * invS;
}

// Unconditional load from a clamped (always-valid) address; zero-pad mask is
// applied separately so the loads can be batched into clauses.
static __device__ __forceinline__ float ldraw(const float* rowp, int x) {
  int xc = x < 0 ? 0 : (x > IMG_W - 1 ? IMG_W - 1 : x);
  return rowp[xc];
}

__global__ __launch_bounds__(256)
void ssim_l1_main(const float* __restrict__ es, const float* __restrict__ ta,
                  float* __restrict__ part) {
  const int lane = threadIdx.x & 31;
  const int wid  = blockIdx.x * 8 + (threadIdx.x >> 5);
  const bool low = lane < 16;
  const int  n   = lane & 15;
  const int  koff = low ? 0 : 2;

  const int plane = wid >> 10;            // 48 planes, 1024 tiles each
  const int t     = wid & 1023;
  const int ybase = (t >> 5) << 4;
  const int xbase = (t & 31) << 4;
  const float* esp = es + (size_t)plane * (IMG_W * IMG_H);
  const float* tap = ta + (size_t)plane * (IMG_W * IMG_H);

  // ---- L1 loads issued first: overlap with the v_exp_f32 weight chain ----
  float le[8], lt[8];
#pragma unroll
  for (int it = 0; it < 8; it++) {
    int idx = it * 32 + lane;
    int off = (ybase + (idx >> 4)) * IMG_W + xbase + (idx & 15);
    le[it] = esp[off];
    lt[it] = tap[off];
  }

  // --- gaussian taps: invS = 1/sum(exp) so taps are normalized like numpy ---
  float s = 0.0f;
#pragma unroll
  for (int k = 0; k < 11; k++) {
    float u = (float)(k - 5);
    s += __expf(-u * u * (1.0f / 4.5f));
  }
  const float invS = 1.0f / s;

  // Band-matrix fragments: value at (K=4c+koff+t, lane&15) = w[K - (lane&15)].
  // Identical per-lane content serves as horizontal-B frag AND vertical-A frag.
  v2f wf[7];
#pragma unroll
  for (int c = 0; c < 7; c++) {
    wf[c].x = gband(4 * c + koff - n, invS);
    wf[c].y = gband(4 * c + koff + 1 - n, invS);
  }

  float lsum = 0.0f;
#pragma unroll
  for (int it = 0; it < 8; it++) lsum += fabsf(le[it] - lt[it]);

  const v8f vzero = {0, 0, 0, 0, 0, 0, 0, 0};
  v8f vacc[5];
#pragma unroll
  for (int q = 0; q < 5; q++) vacc[q] = vzero;

#pragma unroll
  for (int r = 0; r < 2; r++) {
    // ---- horizontal pass: H rows 16r .. 16r+15 (valid rows < 28) ----
    const int rr  = 16 * r + n;                 // H row handled by this lane
    const int y   = ybase - 5 + rr;             // global image row
    const bool yok = (rr < 28) && ((unsigned)y < (unsigned)IMG_H);
    const int yc  = y < 0 ? 0 : (y > IMG_H - 1 ? IMG_H - 1 : y);
    const float* esrow = esp + yc * IMG_W;
    const float* tarow = tap + yc * IMG_W;

    // Load phase: all 28 raw loads issued back-to-back (clause-friendly).
    float ev[14], tv[14];
#pragma unroll
    for (int c = 0; c < 7; c++) {
      const int x0 = xbase - 5 + 4 * c + koff;
      ev[2 * c]     = ldraw(esrow, x0);
      ev[2 * c + 1] = ldraw(esrow, x0 + 1);
      tv[2 * c]     = ldraw(tarow, x0);
      tv[2 * c + 1] = ldraw(tarow, x0 + 1);
    }

    // Compute phase: zero-pad masks, lane-wise products, WMMA accumulate.
    v8f hd[5];
#pragma unroll
    for (int q = 0; q < 5; q++) hd[q] = vzero;

#pragma unroll
    for (int c = 0; c < 7; c++) {
      const int x0 = xbase - 5 + 4 * c + koff;
      const bool ok0 = yok && ((unsigned)x0 < (unsigned)IMG_W);
      const bool ok1 = yok && ((unsigned)(x0 + 1) < (unsigned)IMG_W);
      float e0 = ok0 ? ev[2 * c] : 0.0f;
      float e1 = ok1 ? ev[2 * c + 1] : 0.0f;
      float t0 = ok0 ? tv[2 * c] : 0.0f;
      float t1 = ok1 ? tv[2 * c + 1] : 0.0f;
      v2f af[5];
      af[0].x = e0;      af[0].y = e1;          // es
      af[1].x = t0;      af[1].y = t1;          // ta
      af[2].x = e0 * e0; af[2].y = e1 * e1;     // es*es
      af[3].x = t0 * t0; af[3].y = t1 * t1;     // ta*ta
      af[4].x = e0 * t0; af[4].y = e1 * t1;     // es*ta
#pragma unroll
      for (int q = 0; q < 5; q++) hd[q] = wmma_f32(af[q], wf[c], hd[q]);
    }

    // ---- vertical pass chunks covered by this row block ----
    // r=0 -> K chunks c=0..3 (H rows 0..15); r=1 -> c=4..6 (H rows 16..27)
    // D layout: vgpr i = row i (lanes 0-15) / row 8+i (lanes 16-31), col=lane&15.
    // B frag needs rows {jb,jb+1} in lanes 0-15 and {jb+2,jb+3} in lanes 16-31.
#pragma unroll
    for (int cc = 0; cc < 4; cc++) {
      if (r == 1 && cc == 3) continue;
      const int c  = 4 * r + cc;
      const int jb = 4 * cc;                    // local row base inside hd
      float s0[5], s1[5], b0[5], b1[5];
      if (jb < 8) {                             // rows live in lanes 0-15
#pragma unroll
        for (int q = 0; q < 5; q++) {
          s0[q] = xswap(hd[q][jb + 2]);
          s1[q] = xswap(hd[q][jb + 3]);
        }
#pragma unroll
        for (int q = 0; q < 5; q++) {
          b0[q] = low ? hd[q][jb]     : s0[q];
          b1[q] = low ? hd[q][jb + 1] : s1[q];
        }
      } else {                                  // rows live in lanes 16-31
#pragma unroll
        for (int q = 0; q < 5; q++) {
          s0[q] = xswap(hd[q][jb - 8]);
          s1[q] = xswap(hd[q][jb - 7]);
        }
#pragma unroll
        for (int q = 0; q < 5; q++) {
          b0[q] = low ? s0[q] : hd[q][jb - 6];
          b1[q] = low ? s1[q] : hd[q][jb - 5];
        }
      }
#pragma unroll
      for (int q = 0; q < 5; q++) {
        v2f b; b.x = b0[q]; b.y = b1[q];
        vacc[q] = wmma_f32(wf[c], b, vacc[q]);
      }
    }
  }

  // ---- per-element SSIM map + accumulate ----
  float ssum = 0.0f;
#pragma unroll
  for (int e = 0; e < 8; e++) {
    float mu1 = vacc[0][e], mu2 = vacc[1][e];
    float mu1s = mu1 * mu1, mu2s = mu2 * mu2, mu12 = mu1 * mu2;
    float s1  = vacc[2][e] - mu1s;
    float s2  = vacc[3][e] - mu2s;
    float s12 = vacc[4][e] - mu12;
    float num = (2.0f * mu12 + SSIM_C1) * (2.0f * s12 + SSIM_C2);
    float den = (mu1s + mu2s + SSIM_C1) * (s1 + s2 + SSIM_C2);
    ssum += num / den;
  }

  // ---- deterministic block reduction -> per-block partials ----
  __shared__ float redS[256];
  __shared__ float redL[256];
  redS[threadIdx.x] = ssum;
  redL[threadIdx.x] = lsum;
  __syncthreads();
#pragma unroll
  for (int st = 128; st > 0; st >>= 1) {
    if ((int)threadIdx.x < st) {
      redS[threadIdx.x] += redS[threadIdx.x + st];
      redL[threadIdx.x] += redL[threadIdx.x + st];
    }
    __syncthreads();
  }
  if (threadIdx.x == 0) {
    part[blockIdx.x]            = redS[0];
    part[N_BLOCKS + blockIdx.x] = redL[0];
  }
}

__global__ __launch_bounds__(256)
void ssim_l1_finalize(const float* __restrict__ part, float* __restrict__ out) {
  __shared__ double sa[256];
  __shared__ double sb[256];
  double ss = 0.0, ll = 0.0;
  for (int i = threadIdx.x; i < N_BLOCKS; i += 256) {
    ss += (double)part[i];
    ll += (double)part[N_BLOCKS + i];
  }
  sa[threadIdx.x] = ss;
  sb[threadIdx.x] = ll;
  __syncthreads();
  for (int st = 128; st > 0; st >>= 1) {
    if ((int)threadIdx.x < st) {
      sa[threadIdx.x] += sa[threadIdx.x + st];
      sb[threadIdx.x] += sb[threadIdx.x + st];
    }
    __syncthreads();
  }
  if (threadIdx.x == 0) {
    double ssim_mean = sa[0] / N_ELEMS;
    double l1_mean   = sb[0] / N_ELEMS;
    out[0] = (float)(0.15 * l1_mean);                 // l1_loss
    out[1] = (float)(0.85 * 0.5 * (1.0 - ssim_mean)); // ssim_loss
  }
}

extern "C" void kernel_launch(void* const* d_in, const int* in_sizes, int n_in,
                              void* d_out, int out_size, void* d_ws, size_t ws_size,
                              hipStream_t stream) {
  const float* es = (const float*)d_in[0];
  const float* ta = (const float*)d_in[1];
  float* part = (float*)d_ws;           // 2 * 6144 floats of partials
  float* out  = (float*)d_out;          // [l1_loss, ssim_loss]

  ssim_l1_main<<<N_BLOCKS, 256, 0, stream>>>(es, ta, part);
  ssim_l1_finalize<<<1, 256, 0, stream>>>(part, out);
}